// CRFTagger_32444182954181
// MI455X (gfx1250) — compile-verified
//
#include <hip/hip_runtime.h>

// Problem constants (match reference setup_inputs)
#define B_    32
#define L_    2048
#define D_    512
#define C_    5
#define NROWS (B_ * L_)     // 65536 flattened (b, t) rows
#define EMP   8             // padded emission stride (cols 0..4 valid)
#define SEGLEN 128
#define NSEG  16            // segments per batch row, covering t = 1..2047
#define NEGINF (-1.0e30f)

typedef __attribute__((ext_vector_type(2))) float v2f;
typedef __attribute__((ext_vector_type(8))) float v8f;

// --------------------------------------------------------------------------
// Kernel 1: emissions em[row][c] = Z[row,:] . W[c,:] + b[c] + class_bias[c]
// fp32 WMMA 16x16x4. One wave per 16-row tile, K-loop of 128 WMMA steps.
// A layout (16x4 f32): lane l -> M = l&15, Kpair base = (l>>4)*2, 2 VGPRs.
// B layout (4x16 f32): lane l -> N = l&15, same K pairing (symmetric).
// D layout (16x16 f32): vgpr v, lane l -> row = v + 8*(l>>4), col = l&15.
// --------------------------------------------------------------------------
__global__ __launch_bounds__(256) void em_wmma_kernel(
    const float* __restrict__ Z, const float* __restrict__ W,
    const float* __restrict__ bias, const float* __restrict__ cbias,
    float* __restrict__ em)
{
  // Wt[k/2][n] = (W[n][k], W[n][k+1]); 16 columns, cols >= C_ are zero. 32 KB.
  __shared__ v2f Wt[(D_ / 2) * 16];
  const int tid = threadIdx.x;
  for (int idx = tid; idx < (D_ / 2) * 16; idx += 256) {
    const int kk = idx >> 4;      // K pair index
    const int n  = idx & 15;      // column (N)
    v2f w = {0.0f, 0.0f};
    if (n < C_) {
      w.x = W[n * D_ + 2 * kk];
      w.y = W[n * D_ + 2 * kk + 1];
    }
    Wt[idx] = w;
  }
  __syncthreads();

  const int  wave = tid >> 5;
  const int  lane = tid & 31;
  const int  m    = lane & 15;              // A row / B,D column
  const int  kb   = (lane >> 4) << 1;       // 0 or 2 (K pair select)
  const long tileRow = ((long)blockIdx.x * 8 + wave) * 16;
  const float* __restrict__ zrow = Z + (tileRow + m) * D_;

  v8f acc = {};
  #pragma unroll 4
  for (int k0 = 0; k0 < D_; k0 += 4) {
    v2f a  = *(const v2f*)(zrow + k0 + kb);         // global_load_b64
    v2f bb = Wt[((k0 + kb) >> 1) * 16 + m];         // ds_load_b64
    acc = __builtin_amdgcn_wmma_f32_16x16x4_f32(
        /*neg_a=*/false, a, /*neg_b=*/false, bb,
        /*c_mod=*/(short)0, acc, /*reuse_a=*/false, /*reuse_b=*/false);
  }

  if (m < C_) {
    const float bsum   = bias[m] + cbias[m];
    const long rowBase = tileRow + ((lane >> 4) << 3);
    #pragma unroll
    for (int v = 0; v < 8; ++v)
      em[(rowBase + v) * EMP + m] = acc[v] + bsum;
  }
}

// --------------------------------------------------------------------------
// Kernel 2: per-row length + boundary numerator terms (no atomics).
// numer[b] = start_t[tag0] + em[b,0,tag0] + end_t[tag_{len-1}]
// --------------------------------------------------------------------------
__global__ __launch_bounds__(256) void row_setup_kernel(
    const int* __restrict__ tags, const unsigned char* __restrict__ mask,
    const float* __restrict__ start_t, const float* __restrict__ end_t,
    const float* __restrict__ em, float* __restrict__ numer)
{
  const int b = blockIdx.x;
  const int tid = threadIdx.x;
  __shared__ int red[256];
  int s = 0;
  for (int t = tid; t < L_; t += 256) s += mask[b * L_ + t] ? 1 : 0;
  red[tid] = s;
  __syncthreads();
  for (int off = 128; off > 0; off >>= 1) {
    if (tid < off) red[tid] += red[tid + off];
    __syncthreads();
  }
  if (tid == 0) {
    int len = red[0];
    if (len < 1) len = 1;
    const int t0 = tags[b * L_];
    const int tl = tags[b * L_ + (len - 1)];
    numer[b] = start_t[t0] + em[(long)(b * L_) * EMP + t0] + end_t[tl];
  }
}

// --------------------------------------------------------------------------
// Kernel 3: interior numerator sum (deterministic block reduction per row).
// --------------------------------------------------------------------------
__global__ __launch_bounds__(256) void numer_kernel(
    const int* __restrict__ tags, const unsigned char* __restrict__ mask,
    const float* __restrict__ trans, const float* __restrict__ em,
    float* __restrict__ numer)
{
  const int b = blockIdx.x;
  const int tid = threadIdx.x;
  float s = 0.0f;
  for (int t = 1 + tid; t < L_; t += 256) {
    if (mask[b * L_ + t]) {
      const int pt = tags[b * L_ + t - 1];
      const int ct = tags[b * L_ + t];
      s += trans[pt * C_ + ct] + em[((long)(b * L_ + t)) * EMP + ct];
    }
  }
  __shared__ float red[256];
  red[tid] = s;
  __syncthreads();
  for (int off = 128; off > 0; off >>= 1) {
    if (tid < off) red[tid] += red[tid + off];
    __syncthreads();
  }
  if (tid == 0) numer[b] += red[0];
}

// --------------------------------------------------------------------------
// Kernel 4: per-segment log-semiring 5x5 matrix products (parallel scan).
// One wave per (row, segment). Lane (i,j) keeps row i of the running matrix
// replicated; each step: out = LSE_k(Mrow[k] + T[k][j]) + e_t[j], identity if
// masked, then shuffles redistribute the new row i to its 5 lanes.
// --------------------------------------------------------------------------
__global__ __launch_bounds__(32) void seg_kernel(
    const float* __restrict__ em, const unsigned char* __restrict__ mask,
    const float* __restrict__ trans, float* __restrict__ segM)
{
  const int blk  = blockIdx.x;          // b * NSEG + seg
  const int b    = blk / NSEG;
  const int seg  = blk % NSEG;
  const int lane = threadIdx.x;
  const int id   = lane < 25 ? lane : 24;
  const int i = id / 5, j = id % 5;

  float Mrow[5];
  #pragma unroll
  for (int k = 0; k < 5; ++k) Mrow[k] = (k == i) ? 0.0f : NEGINF;
  float Tcol[5];
  #pragma unroll
  for (int k = 0; k < 5; ++k) Tcol[k] = trans[k * C_ + j];

  const int t0 = 1 + seg * SEGLEN;
  int t1 = t0 + SEGLEN; if (t1 > L_) t1 = L_;
  const long base = (long)b * L_;

  for (int t = t0; t < t1; ++t) {
    const float e  = em[(base + t) * EMP + j];
    const int   mk = mask[base + t];
    const float v0 = Mrow[0] + Tcol[0];
    const float v1 = Mrow[1] + Tcol[1];
    const float v2 = Mrow[2] + Tcol[2];
    const float v3 = Mrow[3] + Tcol[3];
    const float v4 = Mrow[4] + Tcol[4];
    const float mx = fmaxf(fmaxf(fmaxf(v0, v1), fmaxf(v2, v3)), v4);
    const float ss = __expf(v0 - mx) + __expf(v1 - mx) + __expf(v2 - mx) +
                     __expf(v3 - mx) + __expf(v4 - mx);
    float out = mx + __logf(ss) + e;
    out = mk ? out : Mrow[j];            // masked step == identity matrix
    #pragma unroll
    for (int k = 0; k < 5; ++k)
      Mrow[k] = __shfl(out, i * 5 + k, 32);
  }
  if (lane < 25) segM[(long)blk * 32 + lane] = Mrow[j];
}

// --------------------------------------------------------------------------
// Kernel 5: fold 16 segment matrices per row into alpha, log Z, final NLL.
// --------------------------------------------------------------------------
__global__ __launch_bounds__(64) void final_kernel(
    const float* __restrict__ em, const float* __restrict__ segM,
    const float* __restrict__ start_t, const float* __restrict__ end_t,
    const float* __restrict__ numer, float* __restrict__ out)
{
  const int tid = threadIdx.x;
  __shared__ float red[64];
  float contrib = 0.0f;
  if (tid < B_) {
    const int b = tid;
    float alpha[5];
    #pragma unroll
    for (int j = 0; j < 5; ++j)
      alpha[j] = start_t[j] + em[(long)(b * L_) * EMP + j];
    for (int s = 0; s < NSEG; ++s) {
      const float* M = segM + (long)(b * NSEG + s) * 32;
      float na[5];
      #pragma unroll
      for (int j = 0; j < 5; ++j) {
        const float v0 = alpha[0] + M[0 * 5 + j];
        const float v1 = alpha[1] + M[1 * 5 + j];
        const float v2 = alpha[2] + M[2 * 5 + j];
        const float v3 = alpha[3] + M[3 * 5 + j];
        const float v4 = alpha[4] + M[4 * 5 + j];
        const float mx = fmaxf(fmaxf(fmaxf(v0, v1), fmaxf(v2, v3)), v4);
        na[j] = mx + __logf(__expf(v0 - mx) + __expf(v1 - mx) +
                            __expf(v2 - mx) + __expf(v3 - mx) + __expf(v4 - mx));
      }
      #pragma unroll
      for (int j = 0; j < 5; ++j) alpha[j] = na[j];
    }
    float z0 = alpha[0] + end_t[0], z1 = alpha[1] + end_t[1];
    float z2 = alpha[2] + end_t[2], z3 = alpha[3] + end_t[3];
    float z4 = alpha[4] + end_t[4];
    const float mx = fmaxf(fmaxf(fmaxf(z0, z1), fmaxf(z2, z3)), z4);
    const float logz = mx + __logf(__expf(z0 - mx) + __expf(z1 - mx) +
                                   __expf(z2 - mx) + __expf(z3 - mx) +
                                   __expf(z4 - mx));
    contrib = numer[b] - logz;
  }
  red[tid] = contrib;
  __syncthreads();
  for (int off = 32; off > 0; off >>= 1) {
    if (tid < off) red[tid] += red[tid + off];
    __syncthreads();
  }
  if (tid == 0) out[0] = -red[0] / (float)B_;
}

// --------------------------------------------------------------------------
extern "C" void kernel_launch(void* const* d_in, const int* in_sizes, int n_in,
                              void* d_out, int out_size, void* d_ws, size_t ws_size,
                              hipStream_t stream)
{
  (void)in_sizes; (void)n_in; (void)out_size; (void)ws_size;
  const float*         Z    = (const float*)d_in[0];
  const int*           tags = (const int*)d_in[1];
  const unsigned char* mask = (const unsigned char*)d_in[2];  // jnp bool = 1B
  const float*         W    = (const float*)d_in[3];
  const float*         b    = (const float*)d_in[4];
  const float*         cb   = (const float*)d_in[5];
  const float*         st   = (const float*)d_in[6];
  const float*         et   = (const float*)d_in[7];
  const float*         tr   = (const float*)d_in[8];

  float* em    = (float*)d_ws;                    // NROWS * EMP floats (2 MB)
  float* segM  = em + (size_t)NROWS * EMP;        // B_*NSEG*32 floats (64 KB)
  float* numer = segM + (size_t)B_ * NSEG * 32;   // B_ floats

  em_wmma_kernel<<<NROWS / 16 / 8, 256, 0, stream>>>(Z, W, b, cb, em);
  row_setup_kernel<<<B_, 256, 0, stream>>>(tags, mask, st, et, em, numer);
  numer_kernel<<<B_, 256, 0, stream>>>(tags, mask, tr, em, numer);
  seg_kernel<<<B_ * NSEG, 32, 0, stream>>>(em, mask, tr, segM);
  final_kernel<<<1, 64, 0, stream>>>(em, segM, st, et, numer, (float*)d_out);
}